// TrainableSGConvolution_20306605376134
// MI455X (gfx1250) — compile-verified
//
#include <hip/hip_runtime.h>
#include <hip/hip_bf16.h>

#define N_NODES 100000
#define D 64

typedef __attribute__((ext_vector_type(2))) float v2f;
typedef __attribute__((ext_vector_type(8))) float v8f;

// ---------------------------------------------------------------------------
// Zero-fill workspace (float4 stores, grid-stride).
// ---------------------------------------------------------------------------
__global__ __launch_bounds__(256) void zero_f32(float4* __restrict__ p, int n4) {
    int i = blockIdx.x * blockDim.x + threadIdx.x;
    int stride = gridDim.x * blockDim.x;
    float4 z = make_float4(0.f, 0.f, 0.f, 0.f);
    for (; i < n4; i += stride) p[i] = z;
}

// ---------------------------------------------------------------------------
// One SpMM hop: out[row] += w * in[col], COO scatter.
// One wave per edge. Edge metadata (row, col, w) is readfirstlane-forced
// uniform so it lowers to scalar loads (s_load, KMcnt path, broadcast).
// Feature gather: one coalesced float2 per lane = a single 256B wave fetch.
// Scatter: 2 global_atomic_add_f32 per lane; the ~70MB working set is
// resident in MI455X's 192MB L2, so the RMWs resolve in L2.
// ---------------------------------------------------------------------------
__global__ __launch_bounds__(256) void spmm_scatter(const int* __restrict__ ei,
                                                    const float* __restrict__ ew,
                                                    const float* __restrict__ xin,
                                                    float* __restrict__ xout,
                                                    int E) {
    int wave = (int)((blockIdx.x * blockDim.x + threadIdx.x) >> 5);
    int lane = threadIdx.x & 31;
    if (wave >= E) return;                       // wave-uniform guard
    int e = __builtin_amdgcn_readfirstlane(wave); // force SGPR -> scalar loads
    int row = ei[e];                              // edge_index[0][e]
    int col = ei[E + e];                          // edge_index[1][e]
    float w = ew[e];

    const float2* __restrict__ src = (const float2*)(xin + (size_t)col * D);
    float2 v = src[lane];                         // coalesced b64 load
    float* __restrict__ dst = xout + (size_t)row * D + lane * 2;
    atomicAdd(dst + 0, w * v.x);
    atomicAdd(dst + 1, w * v.y);
}

// ---------------------------------------------------------------------------
// out = h @ W^T via V_WMMA_F32_16X16X4_F32.
// One wave owns a 16x64 output row-block: 4 accumulators (one per 16-col
// n-tile), so each A float2 is loaded once and feeds 4 WMMAs. h is read
// exactly once from L2; W (16KB) stays hot in L0.
// A layout (16x4 f32): lanes 0-15 -> M=lane, K=0,1 in v0,v1; lanes 16-31 ->
// K=2,3. B layout (4x16): N=lane%16, same K split. C/D: VGPR r -> M=r / r+8.
// ---------------------------------------------------------------------------
__global__ __launch_bounds__(256) void linear_wmma(const float* __restrict__ h,
                                                   const float* __restrict__ Wm,
                                                   float* __restrict__ out,
                                                   int Mtiles) {
    const int wave = (int)((blockIdx.x * blockDim.x + threadIdx.x) >> 5);
    const int lane = threadIdx.x & 31;
    if (wave >= Mtiles) return;                  // wave-uniform guard
    const int m0   = wave * 16;
    const int l16  = lane & 15;
    const int koff = (lane >> 4) << 1;           // 0 (lanes 0-15) / 2 (16-31)

    const float* __restrict__ arow = h  + (size_t)(m0 + l16) * D + koff;
    const float* __restrict__ brow = Wm + (size_t)l16 * D + koff; // n-tile 0

    v8f c0 = {}, c1 = {}, c2 = {}, c3 = {};
#pragma unroll
    for (int k = 0; k < D; k += 4) {
        v2f a  = *(const v2f*)(arow + k);
        v2f b0 = *(const v2f*)(brow + k);            // n =  0..15
        v2f b1 = *(const v2f*)(brow + 16 * D + k);   // n = 16..31
        v2f b2 = *(const v2f*)(brow + 32 * D + k);   // n = 32..47
        v2f b3 = *(const v2f*)(brow + 48 * D + k);   // n = 48..63
        c0 = __builtin_amdgcn_wmma_f32_16x16x4_f32(false, a, false, b0,
                                                   (short)0, c0, false, false);
        c1 = __builtin_amdgcn_wmma_f32_16x16x4_f32(false, a, false, b1,
                                                   (short)0, c1, false, false);
        c2 = __builtin_amdgcn_wmma_f32_16x16x4_f32(false, a, false, b2,
                                                   (short)0, c2, false, false);
        c3 = __builtin_amdgcn_wmma_f32_16x16x4_f32(false, a, false, b3,
                                                   (short)0, c3, false, false);
    }

    const int mb = m0 + ((lane >> 4) << 3);      // +8 for hi half-wave
    float* __restrict__ orow = out + l16;
#pragma unroll
    for (int r = 0; r < 8; ++r) {
        size_t ro = (size_t)(mb + r) * D;
        orow[ro +  0] = c0[r];
        orow[ro + 16] = c1[r];
        orow[ro + 32] = c2[r];
        orow[ro + 48] = c3[r];
    }
}

// ---------------------------------------------------------------------------
// Launch: zero h1,h2 -> hop1 (x->h1) -> hop2 (h1->h2) -> linear (h2->out)
// ---------------------------------------------------------------------------
extern "C" void kernel_launch(void* const* d_in, const int* in_sizes, int n_in,
                              void* d_out, int out_size, void* d_ws, size_t ws_size,
                              hipStream_t stream) {
    const float* x  = (const float*)d_in[0];   // [N_NODES, 64]
    const int*   ei = (const int*)d_in[1];     // [2, E]
    const float* ew = (const float*)d_in[2];   // [E]
    const float* Wm = (const float*)d_in[3];   // [64, 64]
    float* out = (float*)d_out;

    const int E = in_sizes[2];

    float* h1 = (float*)d_ws;
    float* h2 = h1 + (size_t)N_NODES * D;

    // Zero h1 and h2 (2 * 6.4M floats = 3.2M float4)
    {
        int n4 = (2 * N_NODES * D) / 4;
        zero_f32<<<2048, 256, 0, stream>>>((float4*)d_ws, n4);
    }

    // Hop 1: h1 = A @ x ; Hop 2: h2 = A @ h1  (one wave per edge)
    {
        int blocks = (E + 7) / 8;              // 8 waves per 256-thread block
        spmm_scatter<<<blocks, 256, 0, stream>>>(ei, ew, x, h1, E);
        spmm_scatter<<<blocks, 256, 0, stream>>>(ei, ew, h1, h2, E);
    }

    // out = h2 @ W^T  (one wave per 16-row block, all 4 n-tiles)
    {
        int mtiles = N_NODES / 16;             // 6250, exact
        int blocks = (mtiles + 7) / 8;
        linear_wmma<<<blocks, 256, 0, stream>>>(h2, Wm, out, mtiles);
    }
}